// My_maxpool2_609885356904
// MI455X (gfx1250) — compile-verified
//
#include <hip/hip_runtime.h>
#include <stdint.h>

#define RDX_THREADS 256
#define RDX_ITEMS 16
#define RDX_TILE 4096   // RDX_THREADS * RDX_ITEMS

namespace {
constexpr int   kNR    = 2000000;          // neighbor rows
constexpr int   kC     = 39;               // neighbor cols
constexpr int   kT     = 1000000;          // traindata rows (4 cols)
constexpr int   kPool  = 800000;           // N_POOL
constexpr int   kShy   = kNR - kPool;      // 1200000 "shengyu" rows
constexpr int   kGrp   = kPool / 4;        // 200000 groups of 4
constexpr int   kR     = kGrp + kShy;      // 1400000 output rows
constexpr int   kCout  = 22;               // output cols
constexpr float kMaxMin = -100000.0f;
}

// Order-preserving float->uint transforms (total order, -0 < +0, NaN at ends).
__device__ __forceinline__ unsigned ascBits(float f) {
  unsigned u = __float_as_uint(f);
  return (u & 0x80000000u) ? ~u : (u | 0x80000000u);
}
__device__ __forceinline__ unsigned descBits(float f) { return ~ascBits(f); }

// ---------------- sort 1 init: key = desc(neighbor[:,1]), payload = row ----
__global__ void k_init_sort1(const float* __restrict__ nb,
                             unsigned* __restrict__ key, unsigned* __restrict__ pay) {
  int i = blockIdx.x * blockDim.x + threadIdx.x;
  if (i >= kNR) return;
  key[i] = descBits(nb[(size_t)i * kC + 1]);
  pay[i] = (unsigned)i;
}

// ---------------- radix pass 1/3: per-block digit histogram -----------------
__global__ void k_radix_hist(const unsigned* __restrict__ keyIn, unsigned* __restrict__ hist,
                             int N, int shift, int numBlocks) {
  __shared__ unsigned h[16];
  const int t = threadIdx.x, b = blockIdx.x;
  if (t < 16) h[t] = 0;
  __syncthreads();
  const int base = b * RDX_TILE + t * RDX_ITEMS;
  if (base < N) __builtin_prefetch(keyIn + base, 0, 0);   // global_prefetch_b8
  #pragma unroll
  for (int j = 0; j < RDX_ITEMS; ++j) {
    int gi = base + j;
    if (gi < N) atomicAdd(&h[(keyIn[gi] >> shift) & 15u], 1u);
  }
  __syncthreads();
  if (t < 16) hist[(size_t)t * numBlocks + b] = h[t];     // digit-major layout
}

// ---------------- radix pass 2/3: exclusive scan (single block) -------------
__global__ void k_scan_excl(unsigned* __restrict__ h, int len) {
  __shared__ unsigned tile[RDX_THREADS];
  __shared__ unsigned carry;
  const int t = threadIdx.x;
  if (t == 0) carry = 0;
  __syncthreads();
  for (int base = 0; base < len; base += RDX_THREADS) {
    int i = base + t;
    unsigned v = (i < len) ? h[i] : 0u;
    tile[t] = v;
    __syncthreads();
    for (int off = 1; off < RDX_THREADS; off <<= 1) {
      unsigned add = (t >= off) ? tile[t - off] : 0u;
      __syncthreads();
      tile[t] += add;
      __syncthreads();
    }
    unsigned excl = tile[t] - v + carry;
    if (i < len) h[i] = excl;
    __syncthreads();
    if (t == RDX_THREADS - 1) carry += tile[RDX_THREADS - 1];
    __syncthreads();
  }
}

// ---------------- radix pass 3/3: stable scatter (LDS-staged) ---------------
// Full tiles are staged into LDS with CDNA5 async global->LDS (ASYNCcnt).
__global__ void k_radix_scatter(const unsigned* __restrict__ keyIn, const unsigned* __restrict__ payIn,
                                unsigned* __restrict__ keyOut, unsigned* __restrict__ payOut,
                                const unsigned* __restrict__ hist, int N, int shift, int numBlocks) {
  __shared__ __align__(16) unsigned sKey[RDX_TILE];
  __shared__ __align__(16) unsigned sPay[RDX_TILE];
  __shared__ unsigned Mtx[RDX_THREADS][16];
  __shared__ unsigned dgb[16];
  const int b = blockIdx.x, t = threadIdx.x;
  const int blockBase = b * RDX_TILE;
  int cnt = N - blockBase;
  if (cnt > RDX_TILE) cnt = RDX_TILE;

  if (cnt == RDX_TILE) {
    // each thread stages its own contiguous 16 keys + 16 payloads (64 B each)
    unsigned lK = (unsigned)(uintptr_t)(&sKey[t * RDX_ITEMS]);
    unsigned lP = (unsigned)(uintptr_t)(&sPay[t * RDX_ITEMS]);
    unsigned long long gK = (unsigned long long)(uintptr_t)(keyIn + blockBase + t * RDX_ITEMS);
    unsigned long long gP = (unsigned long long)(uintptr_t)(payIn + blockBase + t * RDX_ITEMS);
    #pragma unroll
    for (int q = 0; q < 4; ++q) {
      asm volatile("global_load_async_to_lds_b128 %0, %1, off"
                   :: "v"(lK + 16u * q), "v"(gK + 16ull * q) : "memory");
      asm volatile("global_load_async_to_lds_b128 %0, %1, off"
                   :: "v"(lP + 16u * q), "v"(gP + 16ull * q) : "memory");
    }
    asm volatile("s_wait_asynccnt 0x0" ::: "memory");
  } else {
    for (int k = t; k < RDX_TILE; k += RDX_THREADS) {
      int gi = blockBase + k;
      sKey[k] = (gi < N) ? keyIn[gi] : 0xFFFFFFFFu;
      sPay[k] = (gi < N) ? payIn[gi] : 0u;
    }
  }
  #pragma unroll
  for (int d = 0; d < 16; ++d) Mtx[t][d] = 0;
  __syncthreads();

  // per-thread digit counts over its in-order 16 items (stability!)
  #pragma unroll
  for (int j = 0; j < RDX_ITEMS; ++j) {
    int li = t * RDX_ITEMS + j;
    if (li < cnt) Mtx[t][(sKey[li] >> shift) & 15u] += 1;
  }
  __syncthreads();
  // exclusive scan over threads per digit + fetch this block's global bases
  if (t < 16) {
    unsigned run = 0;
    for (int r = 0; r < RDX_THREADS; ++r) { unsigned v = Mtx[r][t]; Mtx[r][t] = run; run += v; }
    dgb[t] = hist[(size_t)t * numBlocks + b];
  }
  __syncthreads();
  // stable scatter
  #pragma unroll
  for (int j = 0; j < RDX_ITEMS; ++j) {
    int li = t * RDX_ITEMS + j;
    if (li < cnt) {
      unsigned key = sKey[li];
      unsigned d = (key >> shift) & 15u;
      unsigned pos = dgb[d] + Mtx[t][d];
      Mtx[t][d] = Mtx[t][d] + 1;
      keyOut[pos] = key;
      payOut[pos] = sPay[li];
    }
  }
}

// ------- group-of-4 selection: searchsorted(traindata[:,0]) + row max -------
__global__ void k_best(const float* __restrict__ nb, const float* __restrict__ td,
                       const unsigned* __restrict__ perm1, unsigned* __restrict__ rowlist) {
  int i = blockIdx.x * blockDim.x + threadIdx.x;
  if (i >= kGrp) return;
  int best = -1;
  #pragma unroll
  for (int j = 0; j < 4; ++j) {
    unsigned ridx = perm1[kShy + 4 * i + j];
    const float* row = nb + (size_t)ridx * kC;
    float c0 = row[0];
    int lo = 0, hi = kT;                       // lower_bound on td[:,0] (stride 4)
    while (lo < hi) { int mid = (lo + hi) >> 1; if (td[(size_t)mid * 4] < c0) lo = mid + 1; else hi = mid; }
    int tp = lo; if (tp > kT - 1) tp = kT - 1; // JAX clamps OOB gathers
    float mx = fmaxf(td[(size_t)tp * 4 + 1], fmaxf(td[(size_t)tp * 4 + 2], td[(size_t)tp * 4 + 3]));
    bool cond = ((row[4] != 0.0f) == (mx > kMaxMin));
    if (cond) best = j;
  }
  int k = 4 * i + best;
  if (k < 0) k += kPool;                       // numpy-style negative wrap
  rowlist[i] = perm1[kShy + k];
}

__global__ void k_sheng(const unsigned* __restrict__ perm1, unsigned* __restrict__ rowlist) {
  int s = blockIdx.x * blockDim.x + threadIdx.x;
  if (s >= kShy) return;
  rowlist[kGrp + s] = perm1[s];
}

__global__ void k_init_sort2(const float* __restrict__ nb, const unsigned* __restrict__ rowlist,
                             unsigned* __restrict__ key, unsigned* __restrict__ pay) {
  int i = blockIdx.x * blockDim.x + threadIdx.x;
  if (i >= kR) return;
  unsigned rl = rowlist[i];
  key[i] = descBits(nb[(size_t)rl * kC + 1]);
  pay[i] = rl;
}

__global__ void k_zero(unsigned* p) { if (threadIdx.x == 0 && blockIdx.x == 0) *p = 0u; }

// FR = final row order; keys2 = col (1+m) of sorted 22-col matrix; init sort3
__global__ void k_extract(const float* __restrict__ nb, unsigned* pay,
                          unsigned* __restrict__ FR, float* __restrict__ keys2,
                          unsigned* __restrict__ key, unsigned* __restrict__ lens) {
  int i = blockIdx.x * blockDim.x + threadIdx.x;
  if (i >= kR) return;
  unsigned f = pay[i];
  FR[i] = f;
  float k2 = nb[(size_t)f * kC + 2];
  keys2[i] = k2;
  if (k2 == -1.0f) atomicAdd(lens, 1u);        // order-independent -> deterministic
  key[i] = ascBits(k2);
  pay[i] = (unsigned)i;
}

__global__ void k_build_ksorted(const float* __restrict__ nb, const unsigned* __restrict__ korder,
                                const unsigned* __restrict__ FR, const float* __restrict__ keys2,
                                float* __restrict__ ksorted, float* __restrict__ remap) {
  int i = blockIdx.x * blockDim.x + threadIdx.x;
  if (i >= kR) return;
  unsigned ko = korder[i];
  ksorted[i] = keys2[ko];
  remap[i]   = nb[(size_t)FR[ko] * kC + 0];
}

// final: cols 0..4 passthrough; cols 5..21 = searchsorted remap (rows >= len_s)
__global__ void k_final(const float* __restrict__ nb, const unsigned* __restrict__ FR,
                        const float* __restrict__ ksorted, const float* __restrict__ remap,
                        const unsigned* __restrict__ lens, float* __restrict__ out) {
  long long tid = (long long)blockIdx.x * blockDim.x + threadIdx.x;
  if (tid >= (long long)kR * kCout) return;
  int r = (int)(tid / kCout);
  int c = (int)(tid % kCout);
  size_t base = (size_t)FR[r] * kC;
  if (c == 0) __builtin_prefetch(nb + base, 0, 0);
  float val;
  if (c < 5) {
    val = nb[base + c];
  } else {
    float q = nb[base + 22 + (c - 5)];
    val = q;
    if ((unsigned)r >= *lens) {                 // rows < len_s keep new_nb values
      int lo = 0, hi = kR;                      // ksorted is L2-resident (5.6 MB << 192 MB)
      while (lo < hi) { int mid = (lo + hi) >> 1; if (ksorted[mid] < q) lo = mid + 1; else hi = mid; }
      int pos = lo; if (pos > kR - 1) pos = kR - 1;
      if (q > 0.0f && ksorted[pos] == q) val = remap[pos];
    }
  }
  out[(size_t)r * kCout + c] = val;
}

// ---------------- host driver ----------------
static void run_radix_sort(unsigned* keyA, unsigned* payA, unsigned* keyB, unsigned* payB,
                           unsigned* hist, int N, hipStream_t s) {
  const int nb = (N + RDX_TILE - 1) / RDX_TILE;
  unsigned *ki = keyA, *pi = payA, *ko = keyB, *po = payB;
  for (int pass = 0; pass < 8; ++pass) {        // 8 x 4-bit digits
    int shift = pass * 4;
    k_radix_hist<<<nb, RDX_THREADS, 0, s>>>(ki, hist, N, shift, nb);
    k_scan_excl<<<1, RDX_THREADS, 0, s>>>(hist, 16 * nb);
    k_radix_scatter<<<nb, RDX_THREADS, 0, s>>>(ki, pi, ko, po, hist, N, shift, nb);
    unsigned* tk = ki; ki = ko; ko = tk;
    unsigned* tp = pi; pi = po; po = tp;
  }
  // 8 passes (even) -> results back in keyA/payA
}

extern "C" void kernel_launch(void* const* d_in, const int* in_sizes, int n_in,
                              void* d_out, int out_size, void* d_ws, size_t ws_size,
                              hipStream_t stream) {
  (void)in_sizes; (void)n_in; (void)out_size; (void)ws_size;
  const float* traindata = (const float*)d_in[0];
  const float* nb        = (const float*)d_in[1];
  float* out = (float*)d_out;

  char* w = (char*)d_ws;
  size_t off = 0;
  auto alloc = [&](size_t bytes) -> void* {
    off = (off + 255) & ~(size_t)255;
    void* p = w + off; off += bytes; return p;
  };
  unsigned* keyA    = (unsigned*)alloc((size_t)kNR * 4);
  unsigned* keyB    = (unsigned*)alloc((size_t)kNR * 4);
  unsigned* payA    = (unsigned*)alloc((size_t)kNR * 4);
  unsigned* payB    = (unsigned*)alloc((size_t)kNR * 4);
  unsigned* hist    = (unsigned*)alloc((size_t)16 * 512 * 4);
  unsigned* rowlist = (unsigned*)alloc((size_t)kR * 4);
  unsigned* FR      = (unsigned*)alloc((size_t)kR * 4);
  float*    keys2   = (float*)   alloc((size_t)kR * 4);
  float*    ksorted = (float*)   alloc((size_t)kR * 4);
  float*    remap   = (float*)   alloc((size_t)kR * 4);
  unsigned* lens    = (unsigned*)alloc(256);

  // ---- sort 1: argsort desc over neighbor[:,1] (2M rows) ----
  k_init_sort1<<<(kNR + 255) / 256, 256, 0, stream>>>(nb, keyA, payA);
  run_radix_sort(keyA, payA, keyB, payB, hist, kNR, stream);   // payA = perm1

  // ---- group-of-4 keep + shengyu -> rowlist (1.4M row indices into nb) ----
  k_best <<<(kGrp + 255) / 256, 256, 0, stream>>>(nb, traindata, payA, rowlist);
  k_sheng<<<(kShy + 255) / 256, 256, 0, stream>>>(payA, rowlist);

  // ---- sort 2: argsort desc over selected rows' col 1 ----
  k_init_sort2<<<(kR + 255) / 256, 256, 0, stream>>>(nb, rowlist, keyA, payA);
  run_radix_sort(keyA, payA, keyB, payB, hist, kR, stream);    // payA = FR (row ids)

  // ---- extract FR/keys2, count len_s, init sort 3 (ascending keys2) ----
  k_zero<<<1, 1, 0, stream>>>(lens);
  k_extract<<<(kR + 255) / 256, 256, 0, stream>>>(nb, payA, FR, keys2, keyA, lens);
  run_radix_sort(keyA, payA, keyB, payB, hist, kR, stream);    // payA = korder

  // ---- ksorted + remap gather, then fused searchsorted/remap output ----
  k_build_ksorted<<<(kR + 255) / 256, 256, 0, stream>>>(nb, payA, FR, keys2, ksorted, remap);
  long long total = (long long)kR * kCout;
  k_final<<<(int)((total + 255) / 256), 256, 0, stream>>>(nb, FR, ksorted, remap, lens, out);
}